// HeteroRGATLayer_50500225466598
// MI455X (gfx1250) — compile-verified
//
#include <hip/hip_runtime.h>
#include <hip/hip_bf16.h>

#define N_A 50000
#define N_B 50000
#define NE  800000
#define DIM 128

typedef float v2f __attribute__((ext_vector_type(2)));
typedef float v8f __attribute__((ext_vector_type(8)));

// ---------------------------------------------------------------------------
// GEMM: WH[M x 128] = X[M x 128] @ W^T[128 x 128] + b, via V_WMMA_F32_16X16X4_F32.
// One wave -> one 16x16 output tile. M = 50000 = 16*3125, N = 128 = 16*8,
// K = 128 = 4*32. Grid: 3125 blocks * 8 waves = 25000 tiles exactly.
// A-frag layout (16x4 f32): lane l holds A[M=l%16, K=2*(l/16)+j] in VGPR j.
// B-frag layout (4x16 f32): lane l holds B[K=2*(l/16)+j, N=l%16] in VGPR j.
// C/D layout: lane l, VGPR r holds C[M=8*(l/16)+r, N=l%16].
// ---------------------------------------------------------------------------
__global__ void __launch_bounds__(256)
gemm_wh_kernel(const float* __restrict__ X, const float* __restrict__ W,
               const float* __restrict__ bias, float* __restrict__ WH)
{
    const int wave   = threadIdx.x >> 5;
    const int lane   = threadIdx.x & 31;
    const int tile   = blockIdx.x * 8 + wave;     // 0 .. 24999
    const int tile_m = tile >> 3;                 // 8 tiles along N
    const int tile_n = tile & 7;

    const int half   = lane >> 4;                 // 0 or 1
    const int l16    = lane & 15;
    const int kbase  = half * 2;

    const float* __restrict__ xrow = X + (size_t)(tile_m * 16 + l16) * DIM;
    // B[k][n] = W^T[k][n] = W[n][k]  (W is OUT_DIM x IN_DIM row-major)
    const float* __restrict__ wrow = W + (size_t)(tile_n * 16 + l16) * DIM;

    v8f acc = {};
    #pragma unroll
    for (int k = 0; k < DIM; k += 4) {
        v2f a, b;
        a.x = xrow[k + kbase];  a.y = xrow[k + kbase + 1];
        b.x = wrow[k + kbase];  b.y = wrow[k + kbase + 1];
        acc = __builtin_amdgcn_wmma_f32_16x16x4_f32(
            /*neg_a=*/false, a, /*neg_b=*/false, b,
            /*c_mod=*/(short)0, acc, /*reuse_a=*/false, /*reuse_b=*/false);
    }

    const int out_n  = tile_n * 16 + l16;
    const int out_m0 = tile_m * 16 + half * 8;
    const float bv   = bias[out_n];
    float* __restrict__ outp = WH + (size_t)out_m0 * DIM + out_n;
    #pragma unroll
    for (int r = 0; r < 8; ++r)
        outp[(size_t)r * DIM] = acc[r] + bv;
}

// ---------------------------------------------------------------------------
// Per-row dual dot products: s1[m] = wh[m,:] . a[0:128], s2[m] = wh[m,:] . a[128:256]
// One wave per row, lane handles one float4 chunk, shfl_xor reduction (wave32).
// Grid: 50000 waves = 6250 blocks * 8 waves exactly.
// ---------------------------------------------------------------------------
__global__ void __launch_bounds__(256)
scores_kernel(const float* __restrict__ wh, const float* __restrict__ a,
              float* __restrict__ s1, float* __restrict__ s2)
{
    const int row  = (blockIdx.x * 256 + threadIdx.x) >> 5;
    const int lane = threadIdx.x & 31;

    const float4 v  = ((const float4*)(wh + (size_t)row * DIM))[lane];
    const float4 a1 = ((const float4*)(a))[lane];
    const float4 a2 = ((const float4*)(a + DIM))[lane];

    float d1 = v.x * a1.x + v.y * a1.y + v.z * a1.z + v.w * a1.w;
    float d2 = v.x * a2.x + v.y * a2.y + v.z * a2.z + v.w * a2.w;
    #pragma unroll
    for (int off = 16; off > 0; off >>= 1) {
        d1 += __shfl_xor(d1, off, 32);
        d2 += __shfl_xor(d2, off, 32);
    }
    if (lane == 0) { s1[row] = d1; s2[row] = d2; }
}

// Monotone float -> uint key (order-preserving), memset-0 acts as -inf sentinel.
__device__ __forceinline__ unsigned f2key(float f) {
    unsigned b = __float_as_uint(f);
    return (b & 0x80000000u) ? ~b : (b | 0x80000000u);
}
__device__ __forceinline__ float key2f(unsigned k) {
    unsigned b = (k & 0x80000000u) ? (k ^ 0x80000000u) : ~k;
    return __uint_as_float(b);
}

// ---------------------------------------------------------------------------
// Edge pass A: e = leaky_relu(s_src[src]+s_dst[dst], 0.01) * w; segment max.
// ---------------------------------------------------------------------------
__global__ void __launch_bounds__(256)
edge_max_kernel(const int* __restrict__ src, const int* __restrict__ dst,
                const float* __restrict__ w,
                const float* __restrict__ ssrc, const float* __restrict__ sdst,
                float* __restrict__ ebuf, unsigned* __restrict__ mkey)
{
    const int i = blockIdx.x * 256 + threadIdx.x;   // exact grid, no check
    const int d = dst[i];
    float e = ssrc[src[i]] + sdst[d];
    e = (e > 0.0f) ? e : 0.01f * e;
    e *= w[i];
    ebuf[i] = e;
    atomicMax(&mkey[d], f2key(e));
}

// ---------------------------------------------------------------------------
// Edge pass B: ex = exp(e - m[dst]); segment sum of ex. Overwrites ebuf with ex.
// ---------------------------------------------------------------------------
__global__ void __launch_bounds__(256)
edge_expsum_kernel(const int* __restrict__ dst, float* __restrict__ ebuf,
                   const unsigned* __restrict__ mkey, float* __restrict__ denom)
{
    const int i = blockIdx.x * 256 + threadIdx.x;
    const int d = dst[i];
    const float m = key2f(mkey[d]);   // any dst touched here has a valid key
    const float ex = __expf(ebuf[i] - m);
    ebuf[i] = ex;
    atomicAdd(&denom[d], ex);
}

// ---------------------------------------------------------------------------
// Edge pass C: alpha = ex / max(denom[dst],1e-20); h[dst,:] += alpha*wh_src[src,:]
// One wave per edge; lane handles a float4 (4 scalar f32 atomics).
// Grid: 800000 edges * 32 / 256 = 100000 blocks exactly.
// ---------------------------------------------------------------------------
__global__ void __launch_bounds__(256)
edge_scatter_kernel(const int* __restrict__ src, const int* __restrict__ dst,
                    const float* __restrict__ exbuf, const float* __restrict__ denom,
                    const float* __restrict__ whsrc, float* __restrict__ h)
{
    const int gtid = blockIdx.x * 256 + threadIdx.x;
    const int edge = gtid >> 5;
    const int lane = gtid & 31;

    const int s = src[edge];
    const int d = dst[edge];
    const float alpha = exbuf[edge] / fmaxf(denom[d], 1e-20f);

    const float4 v = ((const float4*)(whsrc + (size_t)s * DIM))[lane];
    float* __restrict__ hp = h + (size_t)d * DIM + lane * 4;
    atomicAdd(hp + 0, alpha * v.x);
    atomicAdd(hp + 1, alpha * v.y);
    atomicAdd(hp + 2, alpha * v.z);
    atomicAdd(hp + 3, alpha * v.w);
}

extern "C" void kernel_launch(void* const* d_in, const int* in_sizes, int n_in,
                              void* d_out, int out_size, void* d_ws, size_t ws_size,
                              hipStream_t stream)
{
    const float* x_a  = (const float*)d_in[0];
    const float* x_b  = (const float*)d_in[1];
    const int*   src1 = (const int*)  d_in[2];
    const int*   dst1 = (const int*)  d_in[3];
    const int*   src2 = (const int*)  d_in[4];
    const int*   dst2 = (const int*)  d_in[5];
    const float* ew1  = (const float*)d_in[6];
    const float* ew2  = (const float*)d_in[7];
    const float* W1   = (const float*)d_in[8];
    const float* b1   = (const float*)d_in[9];
    const float* a1   = (const float*)d_in[10];
    const float* W2   = (const float*)d_in[11];
    const float* b2   = (const float*)d_in[12];
    const float* a2   = (const float*)d_in[13];

    // ---- workspace carve-up (floats) --------------------------------------
    float* ws = (float*)d_ws;
    size_t off = 0;
    float* wh1   = ws + off; off += (size_t)N_A * DIM;   // x_a @ W1^T + b1
    float* wh2s  = ws + off; off += (size_t)N_A * DIM;   // x_a @ W2^T + b2
    float* wh2d  = ws + off; off += (size_t)N_B * DIM;   // x_b @ W2^T + b2
    float* ssrc1 = ws + off; off += N_A;
    float* sdst1 = ws + off; off += N_A;
    float* ssrc2 = ws + off; off += N_A;
    float* sdst2 = ws + off; off += N_B;
    float* junk  = ws + off; off += N_A;                 // unused score slot
    unsigned* mkey1 = (unsigned*)(ws + off); off += N_A; // contiguous zero-init block:
    unsigned* mkey2 = (unsigned*)(ws + off); off += N_B;
    float* den1  = ws + off; off += N_A;
    float* den2  = ws + off; off += N_B;
    float* e1    = ws + off; off += NE;
    float* e2    = ws + off; off += NE;

    float* h_a = (float*)d_out;
    float* h_b = h_a + (size_t)N_A * DIM;

    // ---- init (graph-capture-safe async memsets) --------------------------
    hipMemsetAsync(d_out, 0, (size_t)(N_A + N_B) * DIM * sizeof(float), stream);
    hipMemsetAsync(mkey1, 0, (size_t)(N_A + N_B) * 2 * sizeof(float), stream); // mkey1,mkey2,den1,den2

    // ---- three dense transforms (WMMA f32) --------------------------------
    gemm_wh_kernel<<<3125, 256, 0, stream>>>(x_a, W1, b1, wh1);
    gemm_wh_kernel<<<3125, 256, 0, stream>>>(x_a, W2, b2, wh2s);
    gemm_wh_kernel<<<3125, 256, 0, stream>>>(x_b, W2, b2, wh2d);

    // ---- attention scores -------------------------------------------------
    // r1: wh_src == wh_dst == wh1 -> both halves of a_r1
    scores_kernel<<<6250, 256, 0, stream>>>(wh1,  a1, ssrc1, sdst1);
    // r2: src side uses a_r2[:128], dst side uses a_r2[128:]
    scores_kernel<<<6250, 256, 0, stream>>>(wh2s, a2, ssrc2, junk);
    scores_kernel<<<6250, 256, 0, stream>>>(wh2d, a2, junk,  sdst2);

    // ---- relation 1: edge softmax + scatter into h_a ----------------------
    edge_max_kernel    <<<3125,   256, 0, stream>>>(src1, dst1, ew1, ssrc1, sdst1, e1, mkey1);
    edge_expsum_kernel <<<3125,   256, 0, stream>>>(dst1, e1, mkey1, den1);
    edge_scatter_kernel<<<100000, 256, 0, stream>>>(src1, dst1, e1, den1, wh1, h_a);

    // ---- relation 2: edge softmax + scatter into h_b ----------------------
    edge_max_kernel    <<<3125,   256, 0, stream>>>(src2, dst2, ew2, ssrc2, sdst2, e2, mkey2);
    edge_expsum_kernel <<<3125,   256, 0, stream>>>(dst2, e2, mkey2, den2);
    edge_scatter_kernel<<<100000, 256, 0, stream>>>(src2, dst2, e2, den2, wh2s, h_b);
}